// ApproxNDCGLoss_36532991819990
// MI455X (gfx1250) — compile-verified
//
#include <hip/hip_runtime.h>

// ApproxNDCG loss, B=32, L=2048, TEMPERATURE=1, EPS=1e-10.
// Hot loop: soft_ranks[b,i] = 1 + sum_{j!=i} sigmoid(p_i - p_j)   (B*L^2 work)
// Fused in same loop: hard rank of target t_i for ideal DCG (stable argsort semantics).

#define B_DIM   32
#define L_DIM   2048
#define SPLITS  8
#define TPB     256
#define I_PER_BLOCK (L_DIM / SPLITS)   // 256 -> one i per thread
#define EPS_F   1e-10f
#define LOG2E_F 1.44269504088896340736f

// ---- fast math helpers (lower to v_exp_f32 / v_log_f32, both base-2) ----
__device__ __forceinline__ float fast_exp2(float x) {
#if __has_builtin(__builtin_amdgcn_exp2f)
  return __builtin_amdgcn_exp2f(x);
#else
  return exp2f(x);
#endif
}
__device__ __forceinline__ float fast_log2(float x) {
#if __has_builtin(__builtin_amdgcn_logf)
  return __builtin_amdgcn_logf(x);
#else
  return log2f(x);
#endif
}

// ---- gfx1250 async global->LDS staging (ASYNCcnt path) ----
#if defined(__has_builtin)
#if __has_builtin(__builtin_amdgcn_global_load_async_to_lds_b128)
#define HAVE_ASYNC_LDS 1
#endif
#endif

#if HAVE_ASYNC_LDS
typedef int v4i __attribute__((vector_size(4 * sizeof(int))));
typedef __attribute__((address_space(1))) v4i gv4i;   // global int4
typedef __attribute__((address_space(3))) v4i lv4i;   // LDS int4
#endif

__device__ __forceinline__ void wait_asynccnt0() {
#if __has_builtin(__builtin_amdgcn_s_wait_asynccnt)
  __builtin_amdgcn_s_wait_asynccnt(0);
#else
  asm volatile("s_wait_asynccnt 0" ::: "memory");
#endif
}

__global__ __launch_bounds__(TPB) void ndcg_row_kernel(
    const float* __restrict__ pred,    // [B, L]
    const float* __restrict__ targ,    // [B, L]
    float* __restrict__ partial) {     // [B, SPLITS, 2] (approx, ideal)
  __shared__ __align__(16) float sp[L_DIM];   // predictions row (8 KB)
  __shared__ __align__(16) float st[L_DIM];   // targets row     (8 KB)
  __shared__ float r_approx[TPB];
  __shared__ float r_ideal[TPB];

  const int b   = blockIdx.x;   // 0..31
  const int s   = blockIdx.y;   // 0..SPLITS-1
  const int tid = threadIdx.x;

  const float* __restrict__ prow = pred + (size_t)b * L_DIM;
  const float* __restrict__ trow = targ + (size_t)b * L_DIM;

  // ---- stage the full row into LDS (16 B per lane per op) ----
#if HAVE_ASYNC_LDS
  {
    for (int k = tid; k < L_DIM / 4; k += TPB) {   // 512 float4 chunks, 2/thread
      __builtin_amdgcn_global_load_async_to_lds_b128(
          (gv4i*)(const_cast<float*>(prow) + 4 * k), (lv4i*)(sp + 4 * k), 0, 0);
      __builtin_amdgcn_global_load_async_to_lds_b128(
          (gv4i*)(const_cast<float*>(trow) + 4 * k), (lv4i*)(st + 4 * k), 0, 0);
    }
    wait_asynccnt0();
  }
#else
  for (int k = tid; k < L_DIM / 4; k += TPB) {
    reinterpret_cast<float4*>(sp)[k] = reinterpret_cast<const float4*>(prow)[k];
    reinterpret_cast<float4*>(st)[k] = reinterpret_cast<const float4*>(trow)[k];
  }
#endif
  __syncthreads();

  // ---- per-thread i ----
  const int   i  = s * I_PER_BLOCK + tid;
  const float pi = sp[i];
  const float ti = st[i];

  float sig_acc = 0.0f;   // sum_j sigmoid(p_i - p_j), including j==i (0.5)
  int   rank    = 0;      // #{j: t_j > t_i} + #{j: t_j == t_i && j < i}

  #pragma unroll 4
  for (int j = 0; j < L_DIM; ++j) {
    const float pj = sp[j];                 // uniform index -> LDS broadcast
    const float tj = st[j];
    // sigmoid(x) = 1 / (1 + exp2(-x * log2(e)))   (TEMPERATURE == 1)
    const float e = fast_exp2((pj - pi) * LOG2E_F);
    sig_acc += 1.0f / (1.0f + e);
    rank += ((tj > ti) || (tj == ti && j < i)) ? 1 : 0;
  }

  const float soft_rank = 0.5f + sig_acc;            // 1 + (acc - sigmoid(0))
  const float gain      = fast_exp2(ti) - 1.0f;      // 2^t - 1
  const float approx_c  = gain / fast_log2(soft_rank + 1.0f);
  const float ideal_c   = gain / fast_log2((float)(rank + 2));

  // ---- deterministic block tree reduction ----
  r_approx[tid] = approx_c;
  r_ideal[tid]  = ideal_c;
  __syncthreads();
  for (int off = TPB / 2; off > 0; off >>= 1) {
    if (tid < off) {
      r_approx[tid] += r_approx[tid + off];
      r_ideal[tid]  += r_ideal[tid + off];
    }
    __syncthreads();
  }
  if (tid == 0) {
    partial[(b * SPLITS + s) * 2 + 0] = r_approx[0];
    partial[(b * SPLITS + s) * 2 + 1] = r_ideal[0];
  }
}

__global__ __launch_bounds__(B_DIM) void ndcg_final_kernel(
    const float* __restrict__ partial, float* __restrict__ out) {
  __shared__ float red[B_DIM];
  const int b = threadIdx.x;   // 0..31, one wave32
  float a = 0.0f, d = 0.0f;
  #pragma unroll
  for (int s = 0; s < SPLITS; ++s) {
    a += partial[(b * SPLITS + s) * 2 + 0];
    d += partial[(b * SPLITS + s) * 2 + 1];
  }
  red[b] = 1.0f - a / (d + EPS_F);
  __syncthreads();
  if (b == 0) {
    float m = 0.0f;
    #pragma unroll
    for (int k = 0; k < B_DIM; ++k) m += red[k];   // fixed order -> deterministic
    out[0] = m * (1.0f / (float)B_DIM);
  }
}

extern "C" void kernel_launch(void* const* d_in, const int* in_sizes, int n_in,
                              void* d_out, int out_size, void* d_ws, size_t ws_size,
                              hipStream_t stream) {
  const float* predictions = (const float*)d_in[0];
  const float* targets     = (const float*)d_in[1];
  float* out     = (float*)d_out;
  float* partial = (float*)d_ws;   // B*SPLITS*2 floats = 2 KB

  dim3 grid(B_DIM, SPLITS);
  ndcg_row_kernel<<<grid, TPB, 0, stream>>>(predictions, targets, partial);
  ndcg_final_kernel<<<1, B_DIM, 0, stream>>>(partial, out);
}